// EMDHead_89567247991006
// MI455X (gfx1250) — compile-verified
//
#include <hip/hip_runtime.h>
#include <hip/hip_bf16.h>

// Problem constants (from reference)
constexpr int Bc = 2, Ec = 16, Hc = 160, Wc = 288, Nc = 12, Tc = 2;
constexpr int Pc   = Tc * Hc * Wc;          // 92160 points per (b)
constexpr int PADc = 131072;                // 2^17, sort segment length
constexpr int SEGS = Bc * Nc;               // 24 instances
constexpr int SLICES = 45;                  // P / 2048
constexpr int CHUNK = 2048;                 // LDS bitonic chunk (elements)
constexpr int CHUNKS_PER_SEG = PADc / CHUNK; // 64

typedef __attribute__((ext_vector_type(2))) float v2f;
typedef __attribute__((ext_vector_type(8))) float v8f;

__device__ __forceinline__ v8f wmma_f32x4(v2f a, v2f b, v8f c) {
  // D = A(16x4 f32) * B(4x16 f32) + C(16x16 f32)
  return __builtin_amdgcn_wmma_f32_16x16x4_f32(
      /*neg_a=*/false, a, /*neg_b=*/false, b,
      /*c_mod=*/(short)0, c, /*reuse_a=*/false, /*reuse_b=*/false);
}

// ---------------------------------------------------------------------------
// Kernel Z: zero accumulators + output scalar
// ---------------------------------------------------------------------------
__global__ void zero_kernel(float* S1, float* S2, float* cntg, float* out) {
  int tid = threadIdx.x;
  if (tid < 512) { S1[tid] = 0.f; S2[tid] = 0.f; }
  if (tid < 32)  cntg[tid] = 0.f;
  if (tid == 0)  out[0] = 0.f;
}

// ---------------------------------------------------------------------------
// Kernel A: pack embeddings (B,E,H,W)x2 frames -> embT[b][p][e] (p-major, E contiguous)
// ---------------------------------------------------------------------------
__global__ void pack_kernel(const float* __restrict__ mf1,
                            const float* __restrict__ mf2,
                            float* __restrict__ embT) {
  int idx = blockIdx.x * blockDim.x + threadIdx.x;   // B*T*H*W threads
  if (idx >= Bc * Tc * Hc * Wc) return;
  int hw = idx % (Hc * Wc);
  int bt = idx / (Hc * Wc);
  int t = bt % Tc, b = bt / Tc;
  const float* mf = (t == 0) ? mf1 : mf2;
  int p = t * (Hc * Wc) + hw;
  float* dst = embT + ((size_t)(b * Pc + p)) * 16;
  #pragma unroll
  for (int e = 0; e < 16; ++e)
    dst[e] = mf[(b * Ec + e) * (Hc * Wc) + hw];
}

// ---------------------------------------------------------------------------
// Kernel B: masked first/second moments via WMMA f32 16x16x4.
//   S1[n,e] = sum_p m[n,p]*emb[p,e],  S2[n,e] = sum_p m[n,p]*emb[p,e]^2,
//   cnt[n]  = sum_p m[n,p]
// Branch-free: lanes for padded rows n>=12 load a clamped (valid) row and
// multiply by 0 -> no exec-mask churn around the WMMA K-loop.
// ---------------------------------------------------------------------------
__global__ void stats_kernel(const int* __restrict__ gt,
                             const float* __restrict__ embT,
                             float* __restrict__ S1, float* __restrict__ S2,
                             float* __restrict__ cntg) {
  const int slice = blockIdx.x % SLICES;
  const int b     = blockIdx.x / SLICES;
  const int wave  = threadIdx.x >> 5;
  const int lane  = threadIdx.x & 31;
  const int n     = lane & 15;               // A row (instance)
  const int e     = lane & 15;               // B col (embed channel)
  const bool hi   = (lane >= 16);
  const int pbase = slice * 2048 + wave * 512;
  const int   nClamp = (n < Nc) ? n : (Nc - 1);
  const float nMask  = (n < Nc) ? 1.f : 0.f;
  const int* gtn = gt + (size_t)(b * Nc + nClamp) * Pc;

  v8f acc1 = {}; v8f acc2 = {};
  float cacc = 0.f;
  for (int k = 0; k < 512; k += 4) {
    int p0 = pbase + k + (hi ? 2 : 0);       // even -> 8B-aligned int2
    int2 g = *(const int2*)(gtn + p0);
    v2f a;
    a.x = nMask * (float)g.x;
    a.y = nMask * (float)g.y;
    const float* eb = embT + ((size_t)(b * Pc + p0)) * 16 + e;
    v2f bv; bv.x = eb[0]; bv.y = eb[16];
    v2f b2; b2.x = bv.x * bv.x; b2.y = bv.y * bv.y;
    acc1 = wmma_f32x4(a, bv, acc1);
    acc2 = wmma_f32x4(a, b2, acc2);
    cacc += a.x + a.y;
  }
  // C/D layout: lane<16 vgpr r -> (M=r, N=lane); lane>=16 -> (M=8+r, N=lane-16)
  const int rowBase = hi ? 8 : 0;
  const int col = lane & 15;
  #pragma unroll
  for (int r = 0; r < 8; ++r) {
    atomicAdd(&S1[(b * 16 + rowBase + r) * 16 + col], acc1[r]);
    atomicAdd(&S2[(b * 16 + rowBase + r) * 16 + col], acc2[r]);
  }
  atomicAdd(&cntg[b * 16 + n], cacc);
}

// ---------------------------------------------------------------------------
// Kernel C: finalize stats -> coef[n][0..15]=var, coef[n][16..31]=-2*var*mean,
//           cvec[n] = sum_e var*mean^2
// ---------------------------------------------------------------------------
__global__ void finalize_kernel(const float* __restrict__ S1,
                                const float* __restrict__ S2,
                                const float* __restrict__ cntg,
                                float* __restrict__ coef,
                                float* __restrict__ cvec) {
  int tid = threadIdx.x;                 // 512 threads: (b,n,e)
  int b = tid >> 8, n = (tid >> 4) & 15, e = tid & 15;
  float cnt = cntg[b * 16 + n];
  float mean = S1[(b * 16 + n) * 16 + e] / cnt;
  float var  = (S2[(b * 16 + n) * 16 + e] - cnt * mean * mean) / (cnt - 1.f);
  coef[(b * 16 + n) * 32 + e]      = var;
  coef[(b * 16 + n) * 32 + 16 + e] = -2.f * var * mean;
  __shared__ float sh[512];
  sh[tid] = var * mean * mean;
  __syncthreads();
  if (e == 0) {
    float c = 0.f;
    #pragma unroll
    for (int k = 0; k < 16; ++k) c += sh[tid + k];
    cvec[b * 16 + n] = c;
  }
}

// ---------------------------------------------------------------------------
// Kernel D: d[n,p] = sum_{k<32} coef[n,k]*Bk[p] + c[n] via WMMA (K=32 -> 8 steps),
// then logits -> hinge error -> pack u64 sort key. One wave per 16-point tile.
// ---------------------------------------------------------------------------
__global__ void logits_kernel(const float* __restrict__ embT,
                              const float* __restrict__ coef,
                              const float* __restrict__ cvec,
                              const int* __restrict__ gt,
                              unsigned long long* __restrict__ sortbuf) {
  const int wave = threadIdx.x >> 5;
  const int lane = threadIdx.x & 31;
  const int tile = blockIdx.x * 8 + wave;     // B * (P/16) = 11520 tiles
  const int b  = tile / (Pc / 16);
  const int tp = tile % (Pc / 16);
  const int p0 = tp * 16;
  const int l16 = lane & 15;
  const bool hi = (lane >= 16);

  // A: loop-invariant per-instance coefficients (M=n, K=0..31)
  const float* cf = coef + (b * 16 + l16) * 32;
  v2f A[8];
  #pragma unroll
  for (int j = 0; j < 8; ++j) {
    int kk = j * 4 + (hi ? 2 : 0);
    A[j].x = cf[kk]; A[j].y = cf[kk + 1];
  }
  // B: this lane's point embedding (K row k<16 -> emb^2[k], k>=16 -> emb[k-16])
  const int pt = p0 + l16;
  const float* ep = embT + ((size_t)(b * Pc + pt)) * 16;
  float ev[16], bval[32];
  #pragma unroll
  for (int e = 0; e < 16; ++e) ev[e] = ep[e];
  #pragma unroll
  for (int e = 0; e < 16; ++e) { bval[e] = ev[e] * ev[e]; bval[16 + e] = ev[e]; }

  v8f acc = {};
  #pragma unroll
  for (int j = 0; j < 8; ++j) {
    v2f Bv;
    Bv.x = hi ? bval[j * 4 + 2] : bval[j * 4];
    Bv.y = hi ? bval[j * 4 + 3] : bval[j * 4 + 1];
    acc = wmma_f32x4(A[j], Bv, acc);
  }

  const int rowBase = hi ? 8 : 0;
  #pragma unroll
  for (int r = 0; r < 8; ++r) {
    int n = rowBase + r;
    if (n >= Nc) continue;                    // padded instance rows
    float d = acc[r] + cvec[b * 16 + n];
    float logits = 2.f * __expf(-0.5f * d) - 1.f;
    int m = gt[(size_t)(b * Nc + n) * Pc + pt];
    float err = m ? (1.f - logits) : (1.f + logits);
    unsigned int fb = __float_as_uint(err);
    // order-preserving map: ascending uint == ascending float
    unsigned int key = (fb & 0x80000000u) ? ~fb : (fb | 0x80000000u);
    unsigned long long k64 = ((unsigned long long)key << 32) | (unsigned int)m;
    sortbuf[(size_t)(b * Nc + n) * PADc + pt] = k64;
  }
}

// ---------------------------------------------------------------------------
// Kernel P: pad sort segments [P, PAD) with minimal keys (sort last, weight 0)
// ---------------------------------------------------------------------------
__global__ void pad_kernel(unsigned long long* __restrict__ sortbuf) {
  int idx = blockIdx.x * blockDim.x + threadIdx.x;   // SEGS*(PAD-P)
  int seg = idx / (PADc - Pc);
  int off = idx % (PADc - Pc);
  if (seg >= SEGS) return;
  sortbuf[(size_t)seg * PADc + Pc + off] = 0ull;
}

// ---------------------------------------------------------------------------
// Bitonic sort (descending per 2^17 segment), LDS-staged.
// Direction rule everywhere: desc = ((segPos & size) == 0).
// ---------------------------------------------------------------------------
__device__ __forceinline__ void cmpxchg(unsigned long long& a,
                                        unsigned long long& b, bool desc) {
  bool sw = desc ? (a < b) : (a > b);
  if (sw) { unsigned long long t = a; a = b; b = t; }
}

// Presort: all stages with size <= CHUNK, one 2048-element chunk per workgroup.
__global__ void bitonic_sort_shared(unsigned long long* __restrict__ buf) {
  __shared__ unsigned long long sh[CHUNK];
  const int chunk = blockIdx.x;
  const int seg = chunk / CHUNKS_PER_SEG;
  const int lc  = chunk % CHUNKS_PER_SEG;
  unsigned long long* s = buf + (size_t)seg * PADc + (size_t)lc * CHUNK;
  const int tid = threadIdx.x;               // 1024
  sh[tid]        = s[tid];
  sh[tid + 1024] = s[tid + 1024];
  const int gbase = lc * CHUNK;
  for (int size = 2; size <= CHUNK; size <<= 1) {
    for (int stride = size >> 1; stride >= 1; stride >>= 1) {
      __syncthreads();
      int pos = 2 * tid - (tid & (stride - 1));
      bool desc = (((gbase + pos) & size) == 0);
      unsigned long long a = sh[pos], b = sh[pos + stride];
      cmpxchg(a, b, desc);
      sh[pos] = a; sh[pos + stride] = b;
    }
  }
  __syncthreads();
  s[tid]        = sh[tid];
  s[tid + 1024] = sh[tid + 1024];
}

// Global compare-exchange pass (used only for stride >= CHUNK).
__global__ void bitonic_merge_global(unsigned long long* __restrict__ buf,
                                     int size, int stride) {
  int gtid = blockIdx.x * blockDim.x + threadIdx.x;  // SEGS * PAD/2
  int seg = gtid >> 16;                              // PAD/2 == 65536
  int i   = gtid & 65535;
  unsigned long long* s = buf + (size_t)seg * PADc;
  int pos = 2 * i - (i & (stride - 1));
  bool desc = ((pos & size) == 0);
  unsigned long long a = s[pos], b = s[pos + stride];
  bool sw = desc ? (a < b) : (a > b);
  if (sw) { s[pos] = b; s[pos + stride] = a; }
}

// Merge tail: for one size (> CHUNK), all strides CHUNK/2 .. 1 in LDS.
__global__ void bitonic_merge_shared(unsigned long long* __restrict__ buf,
                                     int size) {
  __shared__ unsigned long long sh[CHUNK];
  const int chunk = blockIdx.x;
  const int seg = chunk / CHUNKS_PER_SEG;
  const int lc  = chunk % CHUNKS_PER_SEG;
  unsigned long long* s = buf + (size_t)seg * PADc + (size_t)lc * CHUNK;
  const int tid = threadIdx.x;               // 1024
  sh[tid]        = s[tid];
  sh[tid + 1024] = s[tid + 1024];
  // size > CHUNK -> direction uniform across the whole chunk
  const bool desc = (((lc * CHUNK) & size) == 0);
  for (int stride = CHUNK / 2; stride >= 1; stride >>= 1) {
    __syncthreads();
    int pos = 2 * tid - (tid & (stride - 1));
    unsigned long long a = sh[pos], b = sh[pos + stride];
    cmpxchg(a, b, desc);
    sh[pos] = a; sh[pos + stride] = b;
  }
  __syncthreads();
  s[tid]        = sh[tid];
  s[tid + 1024] = sh[tid + 1024];
}

// ---------------------------------------------------------------------------
// Kernel F: per-instance Lovász hinge via chunked ballot/popc prefix scan.
// cumN_i = (i+1) - cumP_i for binary labels, so only cumP is scanned.
// ---------------------------------------------------------------------------
__global__ void lovasz_kernel(const unsigned long long* __restrict__ sortbuf,
                              const float* __restrict__ cntg,
                              float* __restrict__ out) {
  const int seg = blockIdx.x;                 // 24 workgroups
  const int b = seg / Nc, n = seg % Nc;
  const float gts = cntg[b * 16 + n];
  const unsigned long long* s = sortbuf + (size_t)seg * PADc;
  __shared__ int   wtot[8];
  __shared__ float red[256];
  const int tid = threadIdx.x, lane = tid & 31, wv = tid >> 5;

  int carry = 0;
  float loss = 0.f;
  for (int base = 0; base < PADc; base += 256) {
    unsigned long long k = s[base + tid];
    int label = (int)(k & 1ull);
    unsigned int u = (unsigned int)(k >> 32);
    float err = (u & 0x80000000u) ? __uint_as_float(u & 0x7FFFFFFFu)
                                  : __uint_as_float(~u);
    unsigned long long bal = __ballot(label);
    int prefIncl = __popcll(bal & ((2ull << lane) - 1ull));  // wave-inclusive
    int wTotal   = __popcll(bal);
    if (lane == 0) wtot[wv] = wTotal;
    __syncthreads();
    int waveOff = 0, chunkTot = 0;
    #pragma unroll
    for (int w = 0; w < 8; ++w) { int t = wtot[w]; if (w < wv) waveOff += t; chunkTot += t; }
    __syncthreads();

    int i = base + tid;
    int cumP = carry + waveOff + prefIncl;     // inclusive prefix of labels
    float inter = gts - (float)cumP;
    float uni   = gts + (float)(i + 1 - cumP);
    float jac   = 1.f - inter / uni;
    float jprev = 0.f;
    if (i > 0) {
      int cp = cumP - label;                   // exclusive prefix
      jprev = 1.f - (gts - (float)cp) / (gts + (float)(i - cp));
    }
    float w = (err > 0.f) ? err : 0.f;         // NaN-safe (padding -> 0)
    loss += w * (jac - jprev);
    carry += chunkTot;
  }
  red[tid] = loss;
  __syncthreads();
  for (int sft = 128; sft > 0; sft >>= 1) {
    if (tid < sft) red[tid] += red[tid + sft];
    __syncthreads();
  }
  if (tid == 0) atomicAdd(out, red[0] * (1.0f / (float)SEGS));
}

// ---------------------------------------------------------------------------
extern "C" void kernel_launch(void* const* d_in, const int* in_sizes, int n_in,
                              void* d_out, int out_size, void* d_ws, size_t ws_size,
                              hipStream_t stream) {
  const float* mf1 = (const float*)d_in[0];
  const float* mf2 = (const float*)d_in[1];
  const int*   gt  = (const int*)d_in[2];
  float* out = (float*)d_out;

  char* ws = (char*)d_ws;
  size_t off = 0;
  float* embT = (float*)(ws + off); off += (size_t)Bc * Pc * 16 * sizeof(float); // 11.8 MB
  float* S1   = (float*)(ws + off); off += 512 * sizeof(float);
  float* S2   = (float*)(ws + off); off += 512 * sizeof(float);
  float* cnt  = (float*)(ws + off); off += 32 * sizeof(float);
  float* coef = (float*)(ws + off); off += 1024 * sizeof(float);
  float* cvec = (float*)(ws + off); off += 32 * sizeof(float);
  off = (off + 255) & ~(size_t)255;
  unsigned long long* sortbuf = (unsigned long long*)(ws + off);                 // 25.2 MB

  zero_kernel<<<1, 512, 0, stream>>>(S1, S2, cnt, out);
  pack_kernel<<<(Bc * Tc * Hc * Wc + 255) / 256, 256, 0, stream>>>(mf1, mf2, embT);
  stats_kernel<<<Bc * SLICES, 128, 0, stream>>>(gt, embT, S1, S2, cnt);
  finalize_kernel<<<1, 512, 0, stream>>>(S1, S2, cnt, coef, cvec);
  logits_kernel<<<(Bc * (Pc / 16)) / 8, 256, 0, stream>>>(embT, coef, cvec, gt, sortbuf);
  pad_kernel<<<(SEGS * (PADc - Pc)) / 256, 256, 0, stream>>>(sortbuf);

  const int ldsBlocks = SEGS * CHUNKS_PER_SEG;       // 1536 chunks
  const int cmpBlocks = (SEGS * (PADc / 2)) / 256;   // 6144

  // All stages size <= 2048 in one LDS kernel.
  bitonic_sort_shared<<<ldsBlocks, 1024, 0, stream>>>(sortbuf);
  // Remaining sizes: global passes for stride >= 2048, LDS tail for the rest.
  for (int size = 2 * CHUNK; size <= PADc; size <<= 1) {
    for (int stride = size >> 1; stride >= CHUNK; stride >>= 1)
      bitonic_merge_global<<<cmpBlocks, 256, 0, stream>>>(sortbuf, size, stride);
    bitonic_merge_shared<<<ldsBlocks, 1024, 0, stream>>>(sortbuf, size);
  }

  lovasz_kernel<<<SEGS, 256, 0, stream>>>(sortbuf, cnt, out);
}